// WeightedGraphConv_69432441307196
// MI455X (gfx1250) — compile-verified
//
#include <hip/hip_runtime.h>

#define N_NODES 50000
#define N_EDGES 1250000
#define D       64          // D_IN == D_OUT == 64
#define ALPHA   0.5f

typedef __attribute__((ext_vector_type(2))) float v2f;
typedef __attribute__((ext_vector_type(8))) float v8f;

// ---------------------------------------------------------------------------
// Kernel 1: zero the accumulator h[N_NODES*D] in workspace (float4 stores)
// ---------------------------------------------------------------------------
__global__ __launch_bounds__(256) void wgc_zero_kernel(float4* __restrict__ p, int n4) {
    int i = blockIdx.x * blockDim.x + threadIdx.x;
    if (i < n4) p[i] = make_float4(0.f, 0.f, 0.f, 0.f);
}

// ---------------------------------------------------------------------------
// Kernel 2: edge aggregation. One wave32 per edge; lane l handles feature
// elements [2l, 2l+1] of the 64-wide row. Gather from feature[src], scale by
// (1-w), atomic-add into h[dst]. Feature table and h both live in the 192MB
// L2, so the random traffic stays on-chip.
// ---------------------------------------------------------------------------
__global__ __launch_bounds__(256) void wgc_edge_agg_kernel(
    const float* __restrict__ feature,
    const float* __restrict__ edge_weight,
    const int*   __restrict__ src,
    const int*   __restrict__ dst,
    float*       __restrict__ h,
    int n_edges)
{
    int tid  = blockIdx.x * blockDim.x + threadIdx.x;
    int e    = tid >> 5;          // one wave per edge
    int lane = tid & 31;
    if (e >= n_edges) return;

    int   s = src[e];
    int   d = dst[e];
    float w = 1.0f - edge_weight[e];

    const float2 f = *reinterpret_cast<const float2*>(
        feature + (size_t)s * D + (size_t)lane * 2);

    float* hp = h + (size_t)d * D + (size_t)lane * 2;
    atomicAdd(hp + 0, f.x * w);
    atomicAdd(hp + 1, f.y * w);
}

// ---------------------------------------------------------------------------
// Kernel 3: out = ALPHA * (h @ W^T + b) via V_WMMA_F32_16X16X4_F32.
// One wave computes one 16x16 output tile; K=64 -> 16 WMMA steps of K=4.
//
// VGPR layouts (ISA 7.12.2, 32-bit):
//   A 16x4:  lane<16 -> M=lane,   K={0,1};  lane>=16 -> M=lane-16, K={2,3}
//   B 4x16:  lane<16 -> N=lane,   K={0,1};  lane>=16 -> N=lane-16, K={2,3}
//            B[k][n] = W[n][k]  (row-major W gives the transpose for free)
//   C/D:     VGPR r  -> M = m0 + r + 8*(lane>>4), N = n0 + (lane&15)
// ---------------------------------------------------------------------------
__global__ __launch_bounds__(256) void wgc_gemm_wmma_kernel(
    const float* __restrict__ h,
    const float* __restrict__ W,
    const float* __restrict__ bias,
    float*       __restrict__ out)
{
    const int lane = threadIdx.x & 31;
    const int wave = threadIdx.x >> 5;                 // 0..7
    const int NT   = D / 16;                           // 4 N-tiles
    const int tiles = (N_NODES / 16) * NT;             // 12500
    int tile = blockIdx.x * 8 + wave;
    if (tile >= tiles) return;                         // whole-wave exit: EXEC stays all-ones

    const int mt = tile / NT;
    const int nt = tile % NT;
    const int m0 = mt * 16;
    const int n0 = nt * 16;
    const int lm = lane & 15;
    const int hi = lane >> 4;                          // 0 or 1

    const float* arow = h + (size_t)(m0 + lm) * D + 2 * hi;
    const float* brow = W + (size_t)(n0 + lm) * D + 2 * hi;

    v8f c = {};
#pragma unroll
    for (int k0 = 0; k0 < D; k0 += 4) {
        v2f a = *reinterpret_cast<const v2f*>(arow + k0);
        v2f b = *reinterpret_cast<const v2f*>(brow + k0);
        // 8 args: (neg_a, A, neg_b, B, c_mod, C, reuse_a, reuse_b)
        c = __builtin_amdgcn_wmma_f32_16x16x4_f32(
            false, a, false, b, (short)0, c, false, false);
    }

    const float bv = bias[n0 + lm];
    float* orow = out + (size_t)(m0 + 8 * hi) * D + n0 + lm;
#pragma unroll
    for (int r = 0; r < 8; ++r) {
        orow[(size_t)r * D] = ALPHA * (c[r] + bv);
    }
}

// ---------------------------------------------------------------------------
// Host-side launcher
// ---------------------------------------------------------------------------
extern "C" void kernel_launch(void* const* d_in, const int* in_sizes, int n_in,
                              void* d_out, int out_size, void* d_ws, size_t ws_size,
                              hipStream_t stream) {
    const float* feature     = (const float*)d_in[0];   // [N_NODES, D]
    const float* edge_weight = (const float*)d_in[1];   // [N_EDGES]
    const float* W           = (const float*)d_in[2];   // [D, D]
    const float* bias        = (const float*)d_in[3];   // [D]
    const int*   src         = (const int*)d_in[4];     // [N_EDGES]
    const int*   dst         = (const int*)d_in[5];     // [N_EDGES]
    float*       out         = (float*)d_out;           // [N_NODES, D]
    float*       h           = (float*)d_ws;            // 50000*64*4 = 12.8 MB scratch

    // 1) zero accumulator
    {
        int n4 = (N_NODES * D) / 4;                     // 800000 float4s
        int blocks = (n4 + 255) / 256;
        wgc_zero_kernel<<<blocks, 256, 0, stream>>>((float4*)h, n4);
    }
    // 2) edge aggregation (one wave per edge)
    {
        long long total = (long long)N_EDGES * 32;
        int blocks = (int)((total + 255) / 256);        // 156250
        wgc_edge_agg_kernel<<<blocks, 256, 0, stream>>>(
            feature, edge_weight, src, dst, h, N_EDGES);
    }
    // 3) WMMA GEMM epilogue
    {
        int tiles = (N_NODES / 16) * (D / 16);          // 12500
        int blocks = (tiles + 7) / 8;                   // 1563
        wgc_gemm_wmma_kernel<<<blocks, 256, 0, stream>>>(h, W, bias, out);
    }
}